// CenterLoss_5746666242138
// MI455X (gfx1250) — compile-verified
//
#include <hip/hip_runtime.h>
#include <hip/hip_bf16.h>

#define NUM_CLASSES 8192
#define FEAT_DIM    2048
#define BATCH       4096
#define TILE_M      16
#define NUM_TILES   (BATCH / TILE_M)            // 256 tiles, one wave each
#define WAVES_PER_BLOCK 8
#define BLOCKS      (NUM_TILES / WAVES_PER_BLOCK) // 32 blocks x 256 threads

typedef __attribute__((ext_vector_type(2))) float v2f;
typedef __attribute__((ext_vector_type(8))) float v8f;

// One wave32 per 16-row tile.
// A (16x4 f32): lanes 0-15 hold row m=lane, K = k0,k0+1 ; lanes 16-31 hold
// row m=lane-16, K = k0+2,k0+3  -> each lane loads a float2 (global_load_b64).
// B (4x16 f32): column n = centers[labels[row0+n]]; lane n (mod 16) loads the
// matching float2 from the gathered center row at the same k offsets.
// D = A*B accumulated over K=2048 gives diag(D)[m] = <x_m, c_{label_m}>.
__global__ __launch_bounds__(256) void center_loss_tile_kernel(
    const float* __restrict__ x, const int* __restrict__ labels,
    const float* __restrict__ centers, float* __restrict__ partial) {
  const int lane  = threadIdx.x & 31;
  const int wave  = threadIdx.x >> 5;
  const int tile  = blockIdx.x * WAVES_PER_BLOCK + wave;
  const int row0  = tile * TILE_M;
  const int m     = lane & 15;
  const int khalf = (lane >> 4) << 1;          // 0 for lanes 0-15, 2 for 16-31
  const int row   = row0 + m;

  const float* __restrict__ xrow = x + (size_t)row * FEAT_DIM + khalf;
  const int    lab               = labels[row];
  const float* __restrict__ crow = centers + (size_t)lab * FEAT_DIM + khalf;

  v8f   acc  = {};
  float xsqp = 0.f, csqp = 0.f;

#pragma unroll 4
  for (int k = 0; k < FEAT_DIM; k += 4) {
    v2f a = *(const v2f*)(xrow + k);
    v2f b = *(const v2f*)(crow + k);
    // 8 args: (neg_a, A, neg_b, B, c_mod, C, reuse_a, reuse_b)
    acc = __builtin_amdgcn_wmma_f32_16x16x4_f32(false, a, false, b,
                                                (short)0, acc, false, false);
    xsqp = fmaf(a.y, a.y, fmaf(a.x, a.x, xsqp));
    csqp = fmaf(b.y, b.y, fmaf(b.x, b.x, csqp));
  }

  // Each lane only covered half the K range (k0..k0+1 vs k0+2..k0+3): fold.
  float xsq = xsqp + __shfl_xor(xsqp, 16);
  float csq = csqp + __shfl_xor(csqp, 16);

  // Diagonal of the 16x16 f32 C/D layout:
  //   m=0..7  -> VGPR m,   lane m
  //   m=8..15 -> VGPR m-8, lane m+16
  float d = 0.f;
#pragma unroll
  for (int r = 0; r < 8; ++r) {
    float e = acc[r];
    if (lane == r || lane == r + 24) d = e;
  }
  d += __shfl_xor(d, 16);                      // lanes 0..15 now hold diag[m]

  float dist = xsq + csq - 2.f * d;            // same formula as reference
  dist = fminf(fmaxf(dist, 1e-12f), 1e12f);
  if (lane >= 16) dist = 0.f;                  // avoid double counting mirror

#pragma unroll
  for (int off = 16; off >= 1; off >>= 1) dist += __shfl_xor(dist, off);
  if (lane == 0) partial[tile] = dist;         // deterministic per-tile slot
}

// Deterministic final reduction of the 256 per-tile partials.
__global__ __launch_bounds__(256) void center_loss_finalize_kernel(
    const float* __restrict__ partial, float* __restrict__ out) {
  __shared__ float smem[WAVES_PER_BLOCK];
  const int t = threadIdx.x;
  float v = partial[t];
#pragma unroll
  for (int off = 16; off >= 1; off >>= 1) v += __shfl_xor(v, off);
  if ((t & 31) == 0) smem[t >> 5] = v;
  __syncthreads();
  if (t == 0) {
    float tot = 0.f;
#pragma unroll
    for (int i = 0; i < WAVES_PER_BLOCK; ++i) tot += smem[i];
    // B*(C-1) zero entries each clamped to 1e-12, then / B  ->  (C-1)*1e-12
    out[0] = tot / (float)BATCH + (float)(NUM_CLASSES - 1) * 1e-12f;
  }
}

extern "C" void kernel_launch(void* const* d_in, const int* in_sizes, int n_in,
                              void* d_out, int out_size, void* d_ws, size_t ws_size,
                              hipStream_t stream) {
  (void)in_sizes; (void)n_in; (void)out_size; (void)ws_size;
  const float* x       = (const float*)d_in[0];
  const int*   labels  = (const int*)d_in[1];
  const float* centers = (const float*)d_in[2];
  float*       partial = (float*)d_ws;         // 256 floats of scratch
  float*       out     = (float*)d_out;

  center_loss_tile_kernel<<<BLOCKS, 256, 0, stream>>>(x, labels, centers, partial);
  center_loss_finalize_kernel<<<1, 256, 0, stream>>>(partial, out);
}